// HR_HLIFLayer2D_30932354466329
// MI455X (gfx1250) — compile-verified
//
#include <hip/hip_runtime.h>
#include <cstdint>
#include <cstddef>

// ---------------------------------------------------------------------------
// HR_HLIFLayer2D: leaky-integrate-and-fire scan over T.
//   v = v*decay + x_t ; s = (v > vth) ; v -= s*vth
// Shapes: x (B,T,C,H,W) f32, params (C,H,W) f32, out (B,T,C,H,W) f32.
// Pure HBM-bound (AI ~0.6 flop/B, 268 MB mandatory traffic -> ~11.5us floor
// at 23.3 TB/s). Plan:
//   * kernel 1: precompute vth/decay tables (512 KB) into d_ws once per call
//   * kernel 2: thread = 4 neurons (float4), v held in VGPRs across T,
//     4-deep async global->LDS ring (ASYNCcnt, NT loads), NT b128 stores.
// ---------------------------------------------------------------------------

typedef float f32x4 __attribute__((ext_vector_type(4)));
// Exact pointee type of the async-LDS builtin's pointer params (per hipcc diag):
typedef int v4i_t __attribute__((__vector_size__(4 * sizeof(int))));

#define B_N   16
#define T_N   32
#define CHW_N 65536              // 64*32*32
#define TPB   256                // 8 wave32 waves per workgroup
#define N4    (CHW_N / 4)        // 16384 float4 columns per (b,t) plane
#define BLK_PER_B (N4 / TPB)     // 64 blocks per batch
#define DEPTH 4                  // async LDS ring slots (3 timesteps ahead)

#if defined(__HIP_DEVICE_COMPILE__) && defined(__gfx1250__) && \
    __has_builtin(__builtin_amdgcn_global_load_async_to_lds_b128) && \
    __has_builtin(__builtin_amdgcn_s_wait_asynccnt)
#define USE_ASYNC_LDS 1
#else
#define USE_ASYNC_LDS 0
#endif

#if USE_ASYNC_LDS
typedef __attribute__((address_space(1))) v4i_t* gv4_ptr;  // global b128
typedef __attribute__((address_space(3))) v4i_t* lv4_ptr;  // LDS b128
#define CPOL_LOAD_NT 1  // TH[2:0]=NT: read-once stream, don't churn L2
#endif

__device__ __forceinline__ float softplus_f(float z) {
  // numerically-stable softplus matching jax.nn.softplus
  return fmaxf(z, 0.0f) + log1pf(expf(-fabsf(z)));
}

__device__ __forceinline__ void compute_params(const f32x4 vr, const f32x4 dr,
                                               f32x4& vth, f32x4& dec) {
#pragma unroll
  for (int i = 0; i < 4; ++i) {
    const float zv = __builtin_fmaf(vr[i], 0.1f, 0.5f);   // VTH_S, VTH_M
    vth[i] = softplus_f(zv) + 0.01f;
    const float zd = __builtin_fmaf(dr[i], 0.1f, 2.0f);   // DECAY_S, DECAY_M
    const float sg = 1.0f / (1.0f + expf(-zd));
    dec[i] = fminf(fmaxf(sg, 0.0f), 0.99f);               // clip to [0, 0.99]
  }
}

// ---- kernel 1: per-neuron learned params -> workspace tables --------------
__global__ void __launch_bounds__(TPB, 1)
lif_param_kernel(const float* __restrict__ vth_raw,
                 const float* __restrict__ decay_raw,
                 float* __restrict__ ws)   // ws[0:CHW)=vth, ws[CHW:2CHW)=decay
{
  const int e0 = (blockIdx.x * TPB + threadIdx.x) * 4;
  const f32x4 vr = *(const f32x4*)(vth_raw  + e0);
  const f32x4 dr = *(const f32x4*)(decay_raw + e0);
  f32x4 vth, dec;
  compute_params(vr, dr, vth, dec);
  *(f32x4*)(ws + e0)         = vth;
  *(f32x4*)(ws + CHW_N + e0) = dec;
}

// ---- kernel 2: the scan ---------------------------------------------------
// PRECOMP=true : p0/p1 are ready-to-use vth/decay tables (from d_ws)
// PRECOMP=false: p0/p1 are raw params, compute in-kernel (ws fallback)
template <bool PRECOMP>
__global__ void __launch_bounds__(TPB, 1)
lif_scan_kernel(const float* __restrict__ x,
                const float* __restrict__ p0,
                const float* __restrict__ p1,
                float* __restrict__ out)
{
  const int tid = threadIdx.x;
  const int b   = blockIdx.x / BLK_PER_B;
  const int blk = blockIdx.x - b * BLK_PER_B;
  const int e0  = (blk * TPB + tid) * 4;   // element offset within C*H*W

  f32x4 vth, dec;
  if (PRECOMP) {
    vth = *(const f32x4*)(p0 + e0);
    dec = *(const f32x4*)(p1 + e0);
  } else {
    compute_params(*(const f32x4*)(p0 + e0), *(const f32x4*)(p1 + e0), vth, dec);
  }

  const float* xb = x   + ((size_t)b * T_N) * CHW_N + e0; // t-stride = CHW_N
  float*       ob = out + ((size_t)b * T_N) * CHW_N + e0;

  f32x4 v = {0.0f, 0.0f, 0.0f, 0.0f};

#if USE_ASYNC_LDS
  // 4-deep async global->LDS ring. Each thread owns a private 16B slot per
  // ring entry; no cross-thread sharing -> no barriers. Async loads complete
  // in order within a wave, so asynccnt<=3 means the oldest (current) slot
  // is resident. The slot being restaged at iteration t held timestep t-1,
  // whose ds_load was already consumed (compiler waits DScnt before use),
  // so the WAR on LDS is safe by program order.
  __shared__ f32x4 stage[DEPTH][TPB];

#pragma unroll
  for (int t = 0; t < DEPTH - 1; ++t)      // prologue: stage t = 0,1,2
    __builtin_amdgcn_global_load_async_to_lds_b128(
        (gv4_ptr)(xb + (size_t)t * CHW_N), (lv4_ptr)&stage[t][tid],
        0, CPOL_LOAD_NT);

#pragma unroll
  for (int t = 0; t < T_N; ++t) {
    if (t + DEPTH - 1 < T_N) {
      __builtin_amdgcn_global_load_async_to_lds_b128(
          (gv4_ptr)(xb + (size_t)(t + DEPTH - 1) * CHW_N),
          (lv4_ptr)&stage[(t + DEPTH - 1) % DEPTH][tid],
          0, CPOL_LOAD_NT);
      if (t + 2 * (DEPTH - 1) < T_N)       // warm GL2 further ahead
        __builtin_prefetch(xb + (size_t)(t + 2 * (DEPTH - 1)) * CHW_N, 0, 1);
    }
    // wait until load t is complete (in-order): asynccnt <= outstanding-1
    {
      const int rem = T_N - 1 - t;         // loads still in flight after t
      if      (rem >= 3) __builtin_amdgcn_s_wait_asynccnt(3);
      else if (rem == 2) __builtin_amdgcn_s_wait_asynccnt(2);
      else if (rem == 1) __builtin_amdgcn_s_wait_asynccnt(1);
      else               __builtin_amdgcn_s_wait_asynccnt(0);
    }

    const f32x4 xt = stage[t % DEPTH][tid];  // ds_load_b128
    f32x4 s4;
#pragma unroll
    for (int i = 0; i < 4; ++i) {
      const float vv = __builtin_fmaf(v[i], dec[i], xt[i]); // leaky integrate
      const float sp = (vv > vth[i]) ? 1.0f : 0.0f;         // fire
      v[i]  = __builtin_fmaf(-sp, vth[i], vv);              // soft reset
      s4[i] = sp;
    }
    // streaming output, never re-read: non-temporal 128-bit store
    __builtin_nontemporal_store(s4, (f32x4*)(ob + (size_t)t * CHW_N));
  }
#else
  // Fallback: register-pipelined direct global loads (still coalesced b128).
  f32x4 xn = *(const f32x4*)xb;
  for (int t = 0; t < T_N; ++t) {
    const f32x4 xt = xn;
    if (t + 1 < T_N) {
      xn = *(const f32x4*)(xb + (size_t)(t + 1) * CHW_N);
      if (t + 2 < T_N)
        __builtin_prefetch(xb + (size_t)(t + 2) * CHW_N, 0, 1);
    }
    f32x4 s4;
#pragma unroll
    for (int i = 0; i < 4; ++i) {
      const float vv = __builtin_fmaf(v[i], dec[i], xt[i]);
      const float sp = (vv > vth[i]) ? 1.0f : 0.0f;
      v[i]  = __builtin_fmaf(-sp, vth[i], vv);
      s4[i] = sp;
    }
    __builtin_nontemporal_store(s4, (f32x4*)(ob + (size_t)t * CHW_N));
  }
#endif
}

extern "C" void kernel_launch(void* const* d_in, const int* in_sizes, int n_in,
                              void* d_out, int out_size, void* d_ws, size_t ws_size,
                              hipStream_t stream) {
  (void)in_sizes; (void)n_in; (void)out_size;
  const float* x         = (const float*)d_in[0];
  const float* vth_raw   = (const float*)d_in[1];
  const float* decay_raw = (const float*)d_in[2];
  float*       out       = (float*)d_out;
  float*       wsf       = (float*)d_ws;

  const dim3 grid(B_N * BLK_PER_B);   // 1024 blocks
  const dim3 block(TPB);              // 256 threads = 8 wave32 waves
  const size_t ws_need = (size_t)2 * CHW_N * sizeof(float);  // 512 KB

  if (d_ws != nullptr && ws_size >= ws_need) {
    lif_param_kernel<<<dim3(N4 / TPB), block, 0, stream>>>(vth_raw, decay_raw, wsf);
    lif_scan_kernel<true><<<grid, block, 0, stream>>>(x, wsf, wsf + CHW_N, out);
  } else {
    lif_scan_kernel<false><<<grid, block, 0, stream>>>(x, vth_raw, decay_raw, out);
  }
}